// KANLayer_40965398069809
// MI455X (gfx1250) — compile-verified
//
#include <hip/hip_runtime.h>
#include <math.h>

// ---- problem constants (from the reference) --------------------------------
#define IN_DIM      256
#define OUT_DIM     256
#define BATCH       512
#define NB          11            // GRID_SIZE + K basis funcs per input
#define SPLINE_COLS (IN_DIM * NB) // 2816
#define KTOT        (SPLINE_COLS + IN_DIM) // 3072 : [bases | silu(x)]
#define SPLITK      4
#define KCHUNK      (KTOT / SPLITK)        // 768

typedef __attribute__((ext_vector_type(2))) float v2f;
typedef __attribute__((ext_vector_type(8))) float v8f;

// ---- cubic B-spline bases on the fixed uniform grid ------------------------
// grid: g[t] = -1.75 + 0.25*t, t = 0..14  (GRID_SIZE=8, K=3, [-1,1])
__device__ __forceinline__ void bspline11(float x, float* bb) {
    float g[15];
#pragma unroll
    for (int t = 0; t < 15; ++t) g[t] = -1.75f + 0.25f * (float)t;
    float b[14];
#pragma unroll
    for (int t = 0; t < 14; ++t) b[t] = (x >= g[t] && x < g[t + 1]) ? 1.0f : 0.0f;
#pragma unroll
    for (int j = 1; j <= 3; ++j) {
        const float r = 1.0f / (0.25f * (float)j);  // uniform knots: both denoms = j*spacing
#pragma unroll
        for (int t = 0; t < 14 - j; ++t)            // in-place: reads old b[t], b[t+1]
            b[t] = (x - g[t]) * r * b[t] + (g[t + j + 1] - x) * r * b[t + 1];
    }
#pragma unroll
    for (int t = 0; t < NB; ++t) bb[t] = b[t];
}

// ---- kernel 1: A'[b, :] = [ bases(x[b,i]) for all i | silu(x[b,i]) ] -------
__global__ void kan_prep_A(const float* __restrict__ x, float* __restrict__ Ap) {
    const int b = blockIdx.x;   // batch row
    const int i = threadIdx.x;  // input feature
    const float xv = x[b * IN_DIM + i];
    float bb[NB];
    bspline11(xv, bb);
    float* row = Ap + (size_t)b * KTOT;
#pragma unroll
    for (int t = 0; t < NB; ++t) row[i * NB + t] = bb[t];
    row[SPLINE_COLS + i] = xv / (1.0f + expf(-xv));   // silu
}

// ---- kernel 2: W'[o, :] = [ univ[o,i]*coef[o,i,:] | resw[o,i] ] ------------
__global__ void kan_prep_W(const float* __restrict__ coef,
                           const float* __restrict__ resw,
                           const float* __restrict__ univ,
                           float* __restrict__ Wp) {
    const int o = blockIdx.x;
    const int i = threadIdx.x;
    const float u = univ[o * IN_DIM + i];
    const float* c = coef + ((size_t)o * IN_DIM + i) * NB;
    float* row = Wp + (size_t)o * KTOT;
#pragma unroll
    for (int t = 0; t < NB; ++t) row[i * NB + t] = u * c[t];
    row[SPLINE_COLS + i] = resw[o * IN_DIM + i];
}

// ---- kernel 3: P[kz] += A'(512x3072) x W'^T(3072x256), fp32 WMMA ----------
// A'/W' are K-major. fp32 16x16x4 operand layout (ISA 7.12.2):
//   A: row m = lane&15, k = 2*(lane>>4) + vgpr      -> contiguous float2/lane
//   B: col n = lane&15, k = 2*(lane>>4) + vgpr      -> contiguous float2/lane
//   C/D: vgpr g holds (m = g + 8*(lane>>4), n = lane&15)
__global__ void kan_gemm_wmma(const float* __restrict__ A,
                              const float* __restrict__ W,
                              float* __restrict__ P) {
    const int lane = threadIdx.x & 31;
    const int wave = threadIdx.x >> 5;      // 4 waves: 4 N-tiles per block
    const int half = lane >> 4;             // 0 / 1
    const int mn   = lane & 15;

    const int m0 = blockIdx.y * 16;
    const int n0 = blockIdx.x * 64 + wave * 16;
    const int kz = blockIdx.z;

    const float* arow = A + (size_t)(m0 + mn) * KTOT + 2 * half;
    const float* wrow = W + (size_t)(n0 + mn) * KTOT + 2 * half;
    const int kbeg = kz * KCHUNK;
    const int kend = kbeg + KCHUNK;

    v8f c = {};
    for (int k = kbeg; k < kend; k += 32) {
        __builtin_prefetch(arow + k + 32, 0, 0);   // global_prefetch_b8: next chunk
        __builtin_prefetch(wrow + k + 32, 0, 0);
#pragma unroll
        for (int kk = 0; kk < 32; kk += 4) {
            v2f a = *(const v2f*)(arow + k + kk);
            v2f b = *(const v2f*)(wrow + k + kk);
            // 8 args: (neg_a, A, neg_b, B, c_mod, C, reuse_a, reuse_b)
            c = __builtin_amdgcn_wmma_f32_16x16x4_f32(false, a, false, b,
                                                      (short)0, c, false, false);
        }
    }

    float* prow = P + ((size_t)kz * BATCH + m0) * OUT_DIM + n0;
#pragma unroll
    for (int g = 0; g < 8; ++g)
        prow[(size_t)(g + 8 * half) * OUT_DIM + mn] = c[g];
}

// ---- kernel 4: fixed-order split-K reduction (deterministic) ---------------
__global__ void kan_reduce(const float* __restrict__ P, float* __restrict__ out) {
    const int idx = blockIdx.x * blockDim.x + threadIdx.x;  // 0..BATCH*OUT_DIM-1
    const int s = BATCH * OUT_DIM;
    out[idx] = (P[idx] + P[s + idx]) + (P[2 * s + idx] + P[3 * s + idx]);
}

// ---- host launcher ---------------------------------------------------------
extern "C" void kernel_launch(void* const* d_in, const int* in_sizes, int n_in,
                              void* d_out, int out_size, void* d_ws, size_t ws_size,
                              hipStream_t stream) {
    const float* x    = (const float*)d_in[0];  // (512, 256)
    const float* coef = (const float*)d_in[1];  // (256, 256, 11)
    const float* resw = (const float*)d_in[2];  // (256, 256)
    const float* univ = (const float*)d_in[3];  // (256, 256)
    float* out = (float*)d_out;                 // (512, 256)

    // workspace layout (floats): A' | W' | split-K partials  (~11.0 MB total)
    float* Ap = (float*)d_ws;                       // 512*3072
    float* Wp = Ap + (size_t)BATCH * KTOT;          // 256*3072
    float* P  = Wp + (size_t)OUT_DIM * KTOT;        // 4*512*256

    kan_prep_A<<<BATCH, IN_DIM, 0, stream>>>(x, Ap);
    kan_prep_W<<<OUT_DIM, IN_DIM, 0, stream>>>(coef, resw, univ, Wp);
    kan_gemm_wmma<<<dim3(OUT_DIM / 64, BATCH / 16, SPLITK), 128, 0, stream>>>(Ap, Wp, P);
    kan_reduce<<<(BATCH * OUT_DIM) / 256, 256, 0, stream>>>(P, out);
}